// MotionBlur_11845519802664
// MI455X (gfx1250) — compile-verified
//
#include <hip/hip_runtime.h>
#include <math.h>

// MotionBlur cross-conv as two banded WMMA matmuls per 16x16 tile (gfx1250).
//
// out(i,j) = p_n*sum x[i-7..i, j] + p_s*sum x[i+1..i+8, j]
//          + p_e*sum x[i, j+1..j+8] + p_w*sum x[i, j-7..j]
// = (Av[16x32] * Xv[32x16]) + (Xh[16x32] * Bh[32x16])   per 16x16 tile,
// evaluated with chained V_WMMA_F32_16X16X4_F32 (fp32 throughout).
//
// Halo windows use base offset -8 (K in 0..31 covers rows/cols i0-8..i0+23):
//   N band: K in [il+1, il+8]   S band: K in [il+9, il+16]
//   W band: K in [jl+1, jl+8]   E band: K in [jl+9, jl+16]
// The -8 base makes horizontal A-operand pairs 8B-aligned -> global_load_b64.
//
// One wave = vertical strip of 8 tiles (amortizes coefficient build). Interior
// tiles: unpredicated base+imm-offset loads. Edge tiles: branchless
// clamp-and-mask (EXEC stays all-ones everywhere, as WMMA requires).

#define IMG_H 512
#define IMG_W 512
#define IMG_B 64
#define TILES_X 32
#define TILES_Y 32
#define TILES_PER_WAVE 8
#define STRIPS_Y (TILES_Y / TILES_PER_WAVE)                 // 4
#define WAVES_PER_BLOCK 8
#define TOTAL_STRIPS (IMG_B * TILES_X * STRIPS_Y)           // 8192
#define BLOCKS (TOTAL_STRIPS / WAVES_PER_BLOCK)             // 1024

typedef __attribute__((ext_vector_type(2))) float v2f;
typedef __attribute__((ext_vector_type(8))) float v8f;

#if defined(__has_builtin)
#if __has_builtin(__builtin_amdgcn_wmma_f32_16x16x4_f32)
#define USE_WMMA_F32 1
#endif
#endif
#ifndef USE_WMMA_F32
#define USE_WMMA_F32 0
#endif

__device__ __forceinline__ void softmax4(const float* __restrict__ logits,
                                         const float* __restrict__ gn,
                                         float& pn, float& ps, float& pe, float& pw) {
    float a0 = logits[0] + gn[0];
    float a1 = logits[1] + gn[1];
    float a2 = logits[2] + gn[2];
    float a3 = logits[3] + gn[3];
    float m = fmaxf(fmaxf(a0, a1), fmaxf(a2, a3));
    float e0 = __expf(a0 - m), e1 = __expf(a1 - m), e2 = __expf(a2 - m), e3 = __expf(a3 - m);
    float inv = 1.0f / (e0 + e1 + e2 + e3);
    pn = e0 * inv; ps = e1 * inv; pe = e2 * inv; pw = e3 * inv;
}

#if USE_WMMA_F32

#define WMMA_F32(A, B, C) \
    __builtin_amdgcn_wmma_f32_16x16x4_f32(false, (A), false, (B), (short)0, (C), false, false)

__global__ __launch_bounds__(32 * WAVES_PER_BLOCK)
void motion_blur_wmma(const float* __restrict__ x,
                      const float* __restrict__ logits,
                      const float* __restrict__ gn,
                      float* __restrict__ out) {
    float pn, ps, pe, pw;
    softmax4(logits, gn, pn, ps, pe, pw);

    const int lane = threadIdx.x & 31;
    const int wave = threadIdx.x >> 5;
    const int hi   = lane >> 4;   // 0 or 1 (half-wave)
    const int n    = lane & 15;   // 0..15

    // ---- Band-matrix coefficient operands, built once per wave ----
    // A layout (16x4 f32): reg v, lane L -> (M = L%16, K = v + 2*(L/16)).
    // B layout (4x16 f32): reg v, lane L -> (K = v + 2*(L/16), N = L%16).
    v2f cv[8];   // vertical coef as A: cv(M=il, K), K = halo row - (i0-8)
    v2f ch[8];   // horizontal coef as B: ch(K, N=jl), K = halo col - (j0-8)
#pragma unroll
    for (int q = 0; q < 8; ++q) {
#pragma unroll
        for (int v = 0; v < 2; ++v) {
            int k = 4 * q + v + 2 * hi;                       // K in 0..31
            float cvv = 0.0f;
            if (k >= n + 1 && k <= n + 8)         cvv = pn;   // N: x[i-7..i]
            else if (k >= n + 9 && k <= n + 16)   cvv = ps;   // S: x[i+1..i+8]
            cv[q][v] = cvv;
            float chv = 0.0f;
            if (k >= n + 1 && k <= n + 8)         chv = pw;   // W: x[j-7..j]
            else if (k >= n + 9 && k <= n + 16)   chv = pe;   // E: x[j+1..j+8]
            ch[q][v] = chv;
        }
    }

    // ---- Strip coordinates: one wave = 8 vertically-stacked tiles ----
    const int strip = blockIdx.x * WAVES_PER_BLOCK + wave;    // 0..8191
    const int bimg  = strip >> 7;                             // / (TILES_X*STRIPS_Y)
    const int rem   = strip & 127;
    const int tx    = rem >> 2;                               // 0..31
    const int sy    = rem & 3;                                // 0..3
    const int j0    = tx * 16;
    const bool hsafe = (tx >= 1) && (tx <= 30);               // cols j0-8..j0+23 in range

    const float* __restrict__ xb = x + (size_t)bimg * (IMG_H * IMG_W);
    float* __restrict__ ob       = out + (size_t)bimg * (IMG_H * IMG_W);

    for (int t = 0; t < TILES_PER_WAVE; ++t) {
        const int ty = sy * TILES_PER_WAVE + t;
        const int i0 = ty * 16;
        const bool vsafe = (ty >= 1) && (ty <= 30);           // rows i0-8..i0+23 in range

        v8f accV = {0.f, 0.f, 0.f, 0.f, 0.f, 0.f, 0.f, 0.f};
        v8f accH = {0.f, 0.f, 0.f, 0.f, 0.f, 0.f, 0.f, 0.f};

        // ---- Vertical term: accV += Av(coef) x Xv(rows i0-8..i0+23, cols j0..j0+15)
        // Xv as B operand: reg v, lane L -> X(i0-8 + 4q + v + 2*hi, j0 + L%16)
        if (vsafe) {
            const float* __restrict__ pv = xb + (i0 - 8 + 2 * hi) * IMG_W + (j0 + n);
#pragma unroll
            for (int q = 0; q < 8; ++q) {
                v2f bv;
                bv[0] = pv[(4 * q + 0) * IMG_W];
                bv[1] = pv[(4 * q + 1) * IMG_W];
                accV = WMMA_F32(cv[q], bv, accV);
            }
        } else {
            const int colv = j0 + n;
#pragma unroll
            for (int q = 0; q < 8; ++q) {
                v2f bv;
#pragma unroll
                for (int v = 0; v < 2; ++v) {
                    int r  = i0 - 8 + 4 * q + v + 2 * hi;
                    int rc = min(max(r, 0), IMG_H - 1);
                    float msk = (r == rc) ? 1.0f : 0.0f;      // SAME zero-padding
                    bv[v] = xb[rc * IMG_W + colv] * msk;
                }
                accV = WMMA_F32(cv[q], bv, accV);
            }
        }

        // ---- Horizontal term: accH += Xh(rows i0..i0+15, cols j0-8..j0+23) x Bh(coef)
        // Xh as A operand: reg v, lane L -> X(i0 + L%16, j0-8 + 4q + v + 2*hi)
        if (hsafe) {
            // Base element index (i0+n)*512 + j0-8+2*hi is even -> 8B aligned,
            // so each chunk's (av[0], av[1]) pair is one global_load_b64.
            const v2f* __restrict__ ph2 =
                (const v2f*)(xb + (i0 + n) * IMG_W + (j0 - 8 + 2 * hi));
#pragma unroll
            for (int q = 0; q < 8; ++q) {
                v2f av = ph2[2 * q];
                accH = WMMA_F32(av, ch[q], accH);
            }
        } else {
            const int rowh = (i0 + n) * IMG_W;
#pragma unroll
            for (int q = 0; q < 8; ++q) {
                v2f av;
#pragma unroll
                for (int v = 0; v < 2; ++v) {
                    int c  = j0 - 8 + 4 * q + v + 2 * hi;
                    int cc = min(max(c, 0), IMG_W - 1);
                    float msk = (c == cc) ? 1.0f : 0.0f;      // SAME zero-padding
                    av[v] = xb[rowh + cc] * msk;
                }
                accH = WMMA_F32(av, ch[q], accH);
            }
        }

        // ---- Store D: reg k, lane L -> out(i0 + k + 8*hi, j0 + L%16) ----
        float* __restrict__ po = ob + (i0 + 8 * hi) * IMG_W + (j0 + n);
#pragma unroll
        for (int k = 0; k < 8; ++k) {
            po[k * IMG_W] = accV[k] + accH[k];
        }
    }
}

#else  // !USE_WMMA_F32 — compile-safe VALU fallback (windowed sums)

__global__ __launch_bounds__(256)
void motion_blur_fallback(const float* __restrict__ x,
                          const float* __restrict__ logits,
                          const float* __restrict__ gn,
                          float* __restrict__ out) {
    float pn, ps, pe, pw;
    softmax4(logits, gn, pn, ps, pe, pw);
    const int total = IMG_B * IMG_H * IMG_W;
    const int stride = gridDim.x * blockDim.x;
    for (int idx = blockIdx.x * blockDim.x + threadIdx.x; idx < total; idx += stride) {
        int j = idx % IMG_W;
        int i = (idx / IMG_W) % IMG_H;
        int b = idx / (IMG_W * IMG_H);
        const float* __restrict__ xb = x + (size_t)b * IMG_H * IMG_W;
        float N = 0.f, S = 0.f, E = 0.f, W = 0.f;
#pragma unroll
        for (int d = 0; d < 8; ++d) {
            int rn = i - 7 + d;  if (rn >= 0)     N += xb[rn * IMG_W + j];
            int rs = i + 1 + d;  if (rs < IMG_H)  S += xb[rs * IMG_W + j];
            int ce = j + 1 + d;  if (ce < IMG_W)  E += xb[i * IMG_W + ce];
            int cw = j - 7 + d;  if (cw >= 0)     W += xb[i * IMG_W + cw];
        }
        out[idx] = pn * N + ps * S + pe * E + pw * W;
    }
}

#endif

extern "C" void kernel_launch(void* const* d_in, const int* in_sizes, int n_in,
                              void* d_out, int out_size, void* d_ws, size_t ws_size,
                              hipStream_t stream) {
    const float* x      = (const float*)d_in[0];   // [64,1,512,512] fp32
    const float* logits = (const float*)d_in[1];   // [1,4] fp32
    const float* gnoise = (const float*)d_in[2];   // [1,4] fp32
    float* out          = (float*)d_out;           // [64,1,512,512] fp32
    (void)in_sizes; (void)n_in; (void)out_size; (void)d_ws; (void)ws_size;

#if USE_WMMA_F32
    motion_blur_wmma<<<BLOCKS, 32 * WAVES_PER_BLOCK, 0, stream>>>(x, logits, gnoise, out);
#else
    motion_blur_fallback<<<2048, 256, 0, stream>>>(x, logits, gnoise, out);
#endif
}